// mycrf_2585570312851
// MI455X (gfx1250) — compile-verified
//
#include <hip/hip_runtime.h>

#define SOS_IDX 0
#define EOS_IDX 1
#define NEGV    -10000.0f
#define KK      128
#define TT      512
#define BB      256
#define BT      16      // batch rows per workgroup
#define LP      132     // padded LDS row stride (floats) to avoid bank conflicts

typedef float v2f __attribute__((ext_vector_type(2)));
typedef float v8f __attribute__((ext_vector_type(8)));

// -------- Kernel 1: one-time precompute of rowmax[j] and E[i][j] = exp(trans[j][i]-rowmax[j])
__global__ void crf_prep(const float* __restrict__ trans,
                         float* __restrict__ E,
                         float* __restrict__ rowmax) {
    int j = threadIdx.x;           // 128 threads
    if (j >= KK) return;
    float mx = -3.4e38f;
    #pragma unroll 4
    for (int i = 0; i < KK; ++i) mx = fmaxf(mx, trans[j * KK + i]);
    rowmax[j] = mx;
    #pragma unroll 4
    for (int i = 0; i < KK; ++i) E[i * KK + j] = __expf(trans[j * KK + i] - mx);
}

// -------- Kernel 2: sequential scan. 16 blocks (one per 16-batch tile) x 256 threads (8 waves).
__global__ void __launch_bounds__(256) crf_scan(const float* __restrict__ h,
                                                const float* __restrict__ trans,
                                                const float* __restrict__ mask,
                                                const float* __restrict__ E,
                                                const float* __restrict__ rowmax,
                                                float* __restrict__ out) {
    __shared__ float s_score[BT][LP];
    __shared__ float s_P[BT][LP];
    __shared__ float s_pm[BT][8];     // per-(row, wave) running max partials
    __shared__ float s_red[BT][16];   // final-reduction scratch
    __shared__ float s_C[BT];
    __shared__ float s_rmx[KK];

    const int tid  = threadIdx.x;
    const int b0   = blockIdx.x * BT;
    const int wave = tid >> 5;          // 0..7, owns columns [wave*16, wave*16+16)
    const int lane = tid & 31;          // wave32
    const int ln15 = lane & 15;         // M row (A) / N col (B,D)
    const int hi   = lane >> 4;         // upper-lane half selector
    const int koff = hi << 1;           // A/B fragment k offset for upper lanes
    const int jbase = wave * 16;
    const int jcol  = jbase + ln15;     // this lane's output column

    // ---- init: score = NEG except SOS column; pm partials; cache rowmax in LDS
    #pragma unroll
    for (int r = 0; r < 8; ++r) {
        int idx = r * 256 + tid;
        int m = idx >> 7, i = idx & 127;
        s_score[m][i] = (i == SOS_IDX) ? 0.0f : NEGV;
    }
    if (tid < 128) {                    // pm[m][w]: wave 0 owns SOS column -> max 0
        int m = tid >> 3, w = tid & 7;
        s_pm[m][w] = (w == 0) ? 0.0f : NEGV;
    }
    if (tid < KK) s_rmx[tid] = rowmax[tid];

    // ---- register prefetch pipeline for h and mask (one step ahead)
    float hv_cur[8], hv_nxt[8], mv_cur[8], mv_nxt[8];
    #pragma unroll
    for (int r = 0; r < 8; ++r) {
        int m = r + hi * 8;
        hv_cur[r] = h[((size_t)(b0 + m) * TT + 0) * KK + jcol];
        mv_cur[r] = mask[(size_t)(b0 + m) * TT + 0];
    }

    for (int t = 0; t < TT; ++t) {
        __syncthreads();                 // prev tail's score/pm writes visible
        // ---- prefetch t+1 (full iteration of latency hiding)
        const int tn = (t + 1 < TT) ? t + 1 : t;
        #pragma unroll
        for (int r = 0; r < 8; ++r) {
            int m = r + hi * 8;
            hv_nxt[r] = h[((size_t)(b0 + m) * TT + tn) * KK + jcol];
            mv_nxt[r] = mask[(size_t)(b0 + m) * TT + tn];
        }
        // ---- stabilizer C_b = max_i score[b,i] from 8 per-wave partials
        if (tid < BT) {
            float c = -3.4e38f;
            #pragma unroll
            for (int u = 0; u < 8; ++u) c = fmaxf(c, s_pm[tid][u]);
            s_C[tid] = c;
        }
        __syncthreads();
        // ---- P = exp(score - C)
        #pragma unroll
        for (int r = 0; r < 8; ++r) {
            int idx = r * 256 + tid;
            int m = idx >> 7, i = idx & 127;
            s_P[m][i] = __expf(s_score[m][i] - s_C[m]);
        }
        __syncthreads();
        // ---- S = P(16xK) x E(KxK) tile via V_WMMA_F32_16X16X4_F32, two acc chains
        v8f acc0 = {};
        v8f acc1 = {};
        #pragma unroll
        for (int k = 0; k < KK; k += 8) {
            v2f a0, b0f, a1, b1f;
            a0.x  = s_P[ln15][k + koff];
            a0.y  = s_P[ln15][k + koff + 1];
            b0f.x = E[(size_t)(k + koff) * KK + jcol];
            b0f.y = E[(size_t)(k + koff + 1) * KK + jcol];
            acc0 = __builtin_amdgcn_wmma_f32_16x16x4_f32(false, a0, false, b0f,
                                                         (short)0, acc0, false, false);
            a1.x  = s_P[ln15][k + 4 + koff];
            a1.y  = s_P[ln15][k + 4 + koff + 1];
            b1f.x = E[(size_t)(k + 4 + koff) * KK + jcol];
            b1f.y = E[(size_t)(k + 4 + koff + 1) * KK + jcol];
            acc1 = __builtin_amdgcn_wmma_f32_16x16x4_f32(false, a1, false, b1f,
                                                         (short)0, acc1, false, false);
        }
        // ---- tail: score_new = h + rowmax + C + log(S); masked select;
        //      cross-lane (16-wide) running max for next step's C
        {
            float rj = s_rmx[jcol];
            #pragma unroll
            for (int r = 0; r < 8; ++r) {
                int m = r + hi * 8;
                float s  = acc0[r] + acc1[r];
                float val = hv_cur[r] + rj + s_C[m] + __logf(fmaxf(s, 1e-30f));
                float ns  = (mv_cur[r] != 0.0f) ? val : s_score[m][jcol];
                s_score[m][jcol] = ns;
                float v = ns;
                v = fmaxf(v, __shfl_xor(v, 1, 32));
                v = fmaxf(v, __shfl_xor(v, 2, 32));
                v = fmaxf(v, __shfl_xor(v, 4, 32));
                v = fmaxf(v, __shfl_xor(v, 8, 32));
                if (ln15 == 0) s_pm[m][wave] = v;
            }
        }
        // ---- rotate prefetch registers
        #pragma unroll
        for (int r = 0; r < 8; ++r) { hv_cur[r] = hv_nxt[r]; mv_cur[r] = mv_nxt[r]; }
    }

    // ---- final: out[b] = logsumexp_j(score[b,j] + trans[EOS,j])
    __syncthreads();
    {
        int m = tid >> 4, seg = tid & 15, base = seg * 8;
        float mx = -3.4e38f;
        #pragma unroll
        for (int u = 0; u < 8; ++u)
            mx = fmaxf(mx, s_score[m][base + u] + trans[EOS_IDX * KK + base + u]);
        s_red[m][seg] = mx;
    }
    __syncthreads();
    if (tid < BT) {
        float c = -3.4e38f;
        #pragma unroll
        for (int u = 0; u < 16; ++u) c = fmaxf(c, s_red[tid][u]);
        s_C[tid] = c;
    }
    __syncthreads();
    {
        int m = tid >> 4, seg = tid & 15, base = seg * 8;
        float sm = 0.0f;
        #pragma unroll
        for (int u = 0; u < 8; ++u)
            sm += __expf(s_score[m][base + u] + trans[EOS_IDX * KK + base + u] - s_C[m]);
        s_red[m][seg] = sm;
    }
    __syncthreads();
    if (tid < BT) {
        float tot = 0.0f;
        #pragma unroll
        for (int u = 0; u < 16; ++u) tot += s_red[tid][u];
        out[b0 + tid] = s_C[tid] + __logf(tot);
    }
}

extern "C" void kernel_launch(void* const* d_in, const int* in_sizes, int n_in,
                              void* d_out, int out_size, void* d_ws, size_t ws_size,
                              hipStream_t stream) {
    (void)in_sizes; (void)n_in; (void)out_size; (void)ws_size;
    const float* h     = (const float*)d_in[0];   // [B,T,K] f32
    const float* trans = (const float*)d_in[1];   // [K,K]   f32
    const float* mask  = (const float*)d_in[2];   // [B,T]   f32
    float* out = (float*)d_out;                   // [B]     f32

    float* E      = (float*)d_ws;                 // [K,K] = 64KB
    float* rowmax = E + KK * KK;                  // [K]

    crf_prep<<<1, 128, 0, stream>>>(trans, E, rowmax);
    crf_scan<<<BB / BT, 256, 0, stream>>>(h, trans, mask, E, rowmax, out);
}